// expLinearTransfo_13958643712057
// MI455X (gfx1250) — compile-verified
//
#include <hip/hip_runtime.h>
#include <stdint.h>

// expm of 1M batched 4x4 homogeneous log-affine matrices (bottom row zero).
// Roofline: 48MB in + 48MB out = 96MB <-> 23.3 TB/s  => ~4.1us floor.
// Compute: ~0.55 GFLOP f32 total (fixed s=2 scaling-and-squaring, Taylor-9
// Horner, branch-free) -- negligible vs HBM time at any plausible f32 rate;
// f64 avoided (RDNA-derived wave32 ISA => likely heavily de-rated f64).
// Data movement: CDNA5 Tensor Data Mover both directions -- one descriptor-
// driven DMA per workgroup each way, OOB clamp in the descriptor handles the
// tail workgroup with zero per-lane guards.

#define WG 256
#define TILE_ELEMS (WG * 12)          // 3072 f32 per workgroup tile
#define TILE_BYTES (TILE_ELEMS * 4)   // 12288 bytes

typedef uint32_t v4u __attribute__((ext_vector_type(4)));
typedef int      v8i __attribute__((ext_vector_type(8)));
typedef int      v4i __attribute__((ext_vector_type(4)));

// Build the two live D# groups for a 1D tile of TILE_ELEMS f32 elements.
// tensor_dim0 = elems_avail => hardware zero-fills OOB reads / drops OOB
// writes, so the tail workgroup needs no guards.
__device__ __forceinline__ void tdm_make_desc(uint64_t gaddr, uint32_t lds_addr,
                                              uint32_t elems_avail,
                                              v4u* g0, v8i* g1) {
  *g0 = (v4u){
      1u,                      // [1:0] count=1 (valid user descriptor)
      lds_addr,                // [63:32] LDS byte address
      (uint32_t)gaddr,         // [95:64] global addr low
      ((uint32_t)(gaddr >> 32) & 0x01FFFFFFu) | (2u << 30)  // addr[56:32]|type=2
  };
  *g1 = (v8i){
      (int)(2u << 16),                          // data_size=2 -> 4-byte elems
      (int)((elems_avail & 0xFFFFu) << 16),     // tensor_dim0[15:0]
      (int)((elems_avail >> 16) | (1u << 16)),  // tensor_dim0[31:16]|tensor_dim1=1
      (int)((uint32_t)TILE_ELEMS << 16),        // tensor_dim1 hi=0 | tile_dim0
      1,                                        // tile_dim1=1, tile_dim2=0
      TILE_ELEMS,                               // tensor_dim0_stride
      0, 0                                      // stride hi / tensor_dim1_stride
  };
}

__device__ __forceinline__ void tdm_load_tile(uint64_t gaddr, uint32_t lds_addr,
                                              uint32_t elems_avail) {
  v4u g0; v8i g1;
  tdm_make_desc(gaddr, lds_addr, elems_avail, &g0, &g1);
  const v4i z = {0, 0, 0, 0};
#if __clang_major__ >= 23
  const v8i z8 = {0, 0, 0, 0, 0, 0, 0, 0};
  __builtin_amdgcn_tensor_load_to_lds(g0, g1, z, z, z8, 0);
#else
  __builtin_amdgcn_tensor_load_to_lds(g0, g1, z, z, 0);
#endif
}

__device__ __forceinline__ void tdm_store_tile(uint64_t gaddr, uint32_t lds_addr,
                                               uint32_t elems_avail) {
  v4u g0; v8i g1;
  tdm_make_desc(gaddr, lds_addr, elems_avail, &g0, &g1);
  const v4i z = {0, 0, 0, 0};
#if __clang_major__ >= 23
  const v8i z8 = {0, 0, 0, 0, 0, 0, 0, 0};
  __builtin_amdgcn_tensor_store_from_lds(g0, g1, z, z, z8, 0);
#else
  __builtin_amdgcn_tensor_store_from_lds(g0, g1, z, z, 0);
#endif
}

// M1 (3x4) times M2 (3x4 with implicit bottom row [0,0,0,1]):
//   M[i][j] = sum_{k<3} M1[i][k]*M2[k][j] + M1[i][3]*(j==3)
__device__ __forceinline__ void affine_mul(const float M1[3][4],
                                           const float M2[3][4],
                                           float M[3][4]) {
#pragma unroll
  for (int i = 0; i < 3; ++i) {
#pragma unroll
    for (int j = 0; j < 4; ++j) {
      float s = M1[i][0] * M2[0][j];
      s = fmaf(M1[i][1], M2[1][j], s);
      s = fmaf(M1[i][2], M2[2][j], s);
      if (j == 3) s += M1[i][3];
      M[i][j] = s;
    }
  }
}

__global__ void __launch_bounds__(WG) expm_affine3x4_kernel(
    const float* __restrict__ in, float* __restrict__ out, int batch) {
  __shared__ float sIn[TILE_ELEMS];
  __shared__ float sOut[TILE_ELEMS];

  const int t = threadIdx.x;

  // ---------- Stage 1: TDM DMA global -> LDS (lane 0 issues; TENSORcnt) ------
  if (t == 0) {
    const uint32_t ldsIn = (uint32_t)(uintptr_t)&sIn[0];
    const uint64_t gaddr =
        (uint64_t)(uintptr_t)in + (uint64_t)blockIdx.x * TILE_BYTES;
    const uint32_t avail = (uint32_t)batch * 12u - blockIdx.x * TILE_ELEMS;
    tdm_load_tile(gaddr, ldsIn, avail);
    __builtin_amdgcn_s_wait_tensorcnt(0);
  }
  __syncthreads();

  // ---------- Stage 2: per-thread expm (f32, branch-free) --------------------
  const int m = blockIdx.x * WG + t;
  if (m < batch) {
    // 48B per thread, 16B-aligned in LDS
    const float4* sp = (const float4*)&sIn[t * 12];
    const float4 r0 = sp[0], r1 = sp[1], r2 = sp[2];

    // B = A / 4  (two squarings at the end)
    float Bm[3][4] = {
        {0.25f * r0.x, 0.25f * r0.y, 0.25f * r0.z, 0.25f * r0.w},
        {0.25f * r1.x, 0.25f * r1.y, 0.25f * r1.z, 0.25f * r1.w},
        {0.25f * r2.x, 0.25f * r2.y, 0.25f * r2.z, 0.25f * r2.w}};

    // P = I; Horner: for k=9..1: P = I + (B*P)/k  ->  P = sum_{j<=9} B^j/j!
    // Truncation ||B||<=~0.2: 0.25^10/10! ~ 3e-13 << f32 eps.
    float P[3][4];
#pragma unroll
    for (int i = 0; i < 3; ++i)
#pragma unroll
      for (int j = 0; j < 4; ++j) P[i][j] = (i == j) ? 1.0f : 0.0f;

#pragma unroll
    for (int k = 9; k >= 1; --k) {
      float M[3][4];
      affine_mul(Bm, P, M);
      const float ck = 1.0f / (float)k;  // constant after full unroll
#pragma unroll
      for (int i = 0; i < 3; ++i)
#pragma unroll
        for (int j = 0; j < 4; ++j)
          P[i][j] = fmaf(ck, M[i][j], (i == j) ? 1.0f : 0.0f);
    }

    // exp(A) = (exp(A/4))^4 : square twice (bottom row stays [0,0,0,1])
#pragma unroll
    for (int s = 0; s < 2; ++s) {
      float M[3][4];
      affine_mul(P, P, M);
#pragma unroll
      for (int i = 0; i < 3; ++i)
#pragma unroll
        for (int j = 0; j < 4; ++j) P[i][j] = M[i][j];
    }

    float4* op = (float4*)&sOut[t * 12];
    op[0] = make_float4(P[0][0], P[0][1], P[0][2], P[0][3]);
    op[1] = make_float4(P[1][0], P[1][1], P[1][2], P[1][3]);
    op[2] = make_float4(P[2][0], P[2][1], P[2][2], P[2][3]);
  }
  __syncthreads();  // ds-wait + barrier: all LDS writes visible to the TDM

  // ---------- Stage 3: TDM DMA LDS -> global (lane 0 issues; TENSORcnt) ------
  // OOB writes beyond tensor_dim0 are dropped by hardware => no tail guards.
  // Wave 0 waits TENSORcnt==0 before s_endpgm, keeping LDS alive until the
  // DMA drains; other waves exit immediately after the barrier.
  if (t == 0) {
    const uint32_t ldsOut = (uint32_t)(uintptr_t)&sOut[0];
    const uint64_t gaddr =
        (uint64_t)(uintptr_t)out + (uint64_t)blockIdx.x * TILE_BYTES;
    const uint32_t avail = (uint32_t)batch * 12u - blockIdx.x * TILE_ELEMS;
    tdm_store_tile(gaddr, ldsOut, avail);
    __builtin_amdgcn_s_wait_tensorcnt(0);
  }
}

extern "C" void kernel_launch(void* const* d_in, const int* in_sizes, int n_in,
                              void* d_out, int out_size, void* d_ws, size_t ws_size,
                              hipStream_t stream) {
  const float* vec = (const float*)d_in[0];  // [B, 12] float32
  // d_in[1] is ndims (==3), baked into the kernel
  float* out = (float*)d_out;                // [B, 3, 4] float32

  const int batch  = in_sizes[0] / 12;
  const int blocks = (batch + WG - 1) / WG;
  expm_affine3x4_kernel<<<dim3(blocks), dim3(WG), 0, stream>>>(vec, out, batch);
}